// RecurrentAdditiveCell_45810121179454
// MI455X (gfx1250) — compile-verified
//
#include <hip/hip_runtime.h>

// CDNA5 / gfx1250, wave32.
// out[b,t,u] = h0[b,u] + 0.5 * cumsum_t(x)[b,t,u],  x: f32[16,4096,1024]
//
// 3-phase segmented scan:
//   K1: raw per-segment sums (stream read of x)
//   K2: in-place exclusive scan of segment sums per column (1 MiB ws)
//   K3: WMMA intra-segment cumsum: 16x16 tile cumsum = L(16x16, lower-tri ones) * X,
//       done as 4 chained V_WMMA_F32_16X16X4_F32 (f32, exact), carry via __shfl.

typedef __attribute__((ext_vector_type(2))) float v2f;
typedef __attribute__((ext_vector_type(8))) float v8f;

#define FACTOR 0.5f
#define B_   16
#define T_   4096
#define U_   1024
#define NSEG 16
#define SEGLEN (T_ / NSEG)   // 256
#define TILES  (SEGLEN / 16) // 16

// ---------------- K1: per-(b,u,seg) raw segment sums -> ws ----------------
__global__ void seg_sum_kernel(const float* __restrict__ x, float* __restrict__ segsum) {
    int idx = blockIdx.x * blockDim.x + threadIdx.x;   // 16*16*1024 = 262144 threads
    int u   = idx & (U_ - 1);
    int rem = idx >> 10;
    int b   = rem & (B_ - 1);
    int seg = rem >> 4;
    const float* p = x + ((size_t)(b * T_ + seg * SEGLEN)) * U_ + u;
    float s0 = 0.f, s1 = 0.f, s2 = 0.f, s3 = 0.f;
    #pragma unroll 4
    for (int t = 0; t < SEGLEN; t += 4) {
        s0 += p[(size_t)(t + 0) * U_];
        s1 += p[(size_t)(t + 1) * U_];
        s2 += p[(size_t)(t + 2) * U_];
        s3 += p[(size_t)(t + 3) * U_];
    }
    segsum[(b * NSEG + seg) * U_ + u] = (s0 + s1) + (s2 + s3);
}

// ---------------- K2: in-place exclusive scan over segments ----------------
__global__ void seg_scan_kernel(float* __restrict__ seg) {
    int idx = blockIdx.x * blockDim.x + threadIdx.x;   // 16*1024 = 16384 threads
    int u = idx & (U_ - 1);
    int b = idx >> 10;
    float run = 0.f;
    #pragma unroll
    for (int s = 0; s < NSEG; ++s) {
        int i = (b * NSEG + s) * U_ + u;
        float v = seg[i];
        seg[i] = run;
        run += v;
    }
}

// ---------------- K3: WMMA tile cumsum + carry + writeback ----------------
// One wave per (b, u_tile, seg): 16 * 64 * 16 = 16384 waves.
// A (16x4 f32, lower-tri-ones slice): lanes 0-15 -> M=lane, K=4kb+{0,1};
//                                     lanes 16-31 -> M=lane-16, K=4kb+{2,3}.
// B (4x16 f32): lanes 0-15 -> N=lane, rows K=4kb+{0,1} in VGPR .x/.y;
//               lanes 16-31 -> N=lane-16, rows K=4kb+{2,3}.
// C/D (16x16 f32): VGPR v: lanes 0-15 -> (M=v, N=lane); lanes 16-31 -> (M=v+8, N=lane-16).
__global__ void wmma_scan_kernel(const float* __restrict__ x,
                                 const float* __restrict__ h0,
                                 const float* __restrict__ segoff,
                                 float* __restrict__ out) {
    const int lane  = threadIdx.x & 31;
    const int wave  = threadIdx.x >> 5;
    const int gid   = blockIdx.x * (blockDim.x >> 5) + wave;  // 0..16383
    const int u_tile = gid & 63;
    const int rem    = gid >> 6;
    const int b      = rem & 15;
    const int seg    = rem >> 4;

    const int row   = lane & 15;
    const int khalf = (lane >> 4) << 1;          // 0 (lo half) or 2 (hi half)
    const int u     = (u_tile << 4) + row;

    // Constant A operands: L[t][k] = (k <= t) (inclusive cumsum)
    v2f A[4];
    #pragma unroll
    for (int kb = 0; kb < 4; ++kb) {
        int k0 = 4 * kb + khalf;
        A[kb].x = (k0     <= row) ? 1.0f : 0.0f;
        A[kb].y = (k0 + 1 <= row) ? 1.0f : 0.0f;
    }

    float carry   = segoff[(b * NSEG + seg) * U_ + u];  // exclusive prefix (unscaled)
    const float h = h0[b * U_ + u];

    const size_t colbase = ((size_t)(b * T_ + seg * SEGLEN)) * U_ + u;
    const float* xp = x + colbase;
    float* op = out + colbase + (size_t)(((lane >> 4) << 3)) * U_; // upper half stores rows +8

    // Prime first tile's B operands
    v2f Bc[4], Bn[4];
    #pragma unroll
    for (int kb = 0; kb < 4; ++kb) {
        int r0 = 4 * kb + khalf;
        Bc[kb].x = xp[(size_t)r0 * U_];
        Bc[kb].y = xp[(size_t)(r0 + 1) * U_];
    }

    for (int tile = 0; tile < TILES; ++tile) {
        const float* xn = xp + (size_t)16 * U_;
        if (tile + 1 < TILES) {           // prefetch next tile before the matrix work
            #pragma unroll
            for (int kb = 0; kb < 4; ++kb) {
                int r0 = 4 * kb + khalf;
                Bn[kb].x = xn[(size_t)r0 * U_];
                Bn[kb].y = xn[(size_t)(r0 + 1) * U_];
            }
        }

        // D = L * X via 4 chained 16x16x4 f32 WMMAs
        v8f acc = {};
        #pragma unroll
        for (int kb = 0; kb < 4; ++kb) {
            acc = __builtin_amdgcn_wmma_f32_16x16x4_f32(
                false, A[kb], false, Bc[kb], (short)0, acc, false, false);
        }

        // out = h0 + 0.5 * (carry + intra-tile cumsum)
        #pragma unroll
        for (int v = 0; v < 8; ++v)
            op[(size_t)v * U_] = h + FACTOR * (carry + acc[v]);

        // tile total = cumsum row 15 = acc[7] held by lanes 16..31
        carry += __shfl(acc[7], 16 + row, 32);

        xp = xn;
        op += (size_t)16 * U_;
        if (tile + 1 < TILES) {
            #pragma unroll
            for (int kb = 0; kb < 4; ++kb) Bc[kb] = Bn[kb];
        }
    }
}

extern "C" void kernel_launch(void* const* d_in, const int* in_sizes, int n_in,
                              void* d_out, int out_size, void* d_ws, size_t ws_size,
                              hipStream_t stream) {
    const float* x  = (const float*)d_in[0];   // [16,4096,1024] f32
    const float* h0 = (const float*)d_in[1];   // [16,1024] f32
    float* out      = (float*)d_out;           // [16,4096,1024] f32
    float* segbuf   = (float*)d_ws;            // 16*16*1024 f32 = 1 MiB

    // K1: 262144 threads
    seg_sum_kernel<<<(B_ * NSEG * U_) / 256, 256, 0, stream>>>(x, segbuf);
    // K2: 16384 threads
    seg_scan_kernel<<<(B_ * U_) / 256, 256, 0, stream>>>(segbuf);
    // K3: 16384 waves, 8 waves (256 threads) per block
    wmma_scan_kernel<<<(B_ * 64 * NSEG) / 8, 256, 0, stream>>>(x, h0, segbuf, out);
}